// LowRankRNN_48369921687955
// MI455X (gfx1250) — compile-verified
//
#include <hip/hip_runtime.h>
#include <math.h>

typedef float v2f __attribute__((ext_vector_type(2)));
typedef float v4f __attribute__((ext_vector_type(4)));
typedef float v8f __attribute__((ext_vector_type(8)));

#define ALPHA 0.1f
#define NB    32      // batch
#define NT    2048    // timesteps
#define DIN   128     // input dim
#define NH    512     // hidden dim

// ---------------------------------------------------------------------------
// Kernel 1: xp[bt, h] = sum_k x[bt, k] * I[h, k]   (x @ I^T), f32 WMMA 16x16x4
// One wave (32 lanes) computes one 16x16 tile of xp; K=128 -> 32 WMMA steps.
// A (16x4 f32): lanes 0-15 -> rows M=0..15, K={0,1}; lanes 16-31 -> K={2,3}.
// B (4x16 f32): lanes give column N, half-wave gives K pair (symmetric).
// C/D (16x16 f32, 8 VGPRs): c[v] -> row M = v + 8*(lane/16), col N = lane%16.
// ---------------------------------------------------------------------------
__global__ __launch_bounds__(32)
void lrnn_xproj_wmma(const float* __restrict__ X,   // [NB*NT, DIN]
                     const float* __restrict__ I,   // [NH, DIN]
                     float* __restrict__ XP) {      // [NB*NT, NH]
  const int lane = threadIdx.x;            // 0..31, wave32
  const int m0   = blockIdx.x * 16;        // row tile in flattened B*T
  const int n0   = blockIdx.y * 16;        // col tile in H
  const int half = lane >> 4;              // 0 or 1
  const int l16  = lane & 15;
  const int arow = m0 + l16;               // A-matrix row supplied by this lane
  const int bcol = n0 + l16;               // B-matrix col supplied by this lane

  v8f c = {};
#pragma unroll
  for (int k0 = 0; k0 < DIN; k0 += 4) {
    const int k = k0 + half * 2;           // this lane's K pair
    v2f a = *(const v2f*)(X + (size_t)arow * DIN + k);   // one b64 load
    v2f b = *(const v2f*)(I + (size_t)bcol * DIN + k);   // one b64 load
    c = __builtin_amdgcn_wmma_f32_16x16x4_f32(
        /*neg_a=*/false, a, /*neg_b=*/false, b,
        /*c_mod=*/(short)0, c, /*reuse_a=*/false, /*reuse_b=*/false);
  }

#pragma unroll
  for (int v = 0; v < 8; ++v) {
    const int row = m0 + v + 8 * half;
    XP[(size_t)row * NH + n0 + l16] = c[v];
  }
}

// ---------------------------------------------------------------------------
// Kernel 2: sequential scan. J = m n^T is rank-2, so
//   tanh(h) @ J^T = (tanh(h) @ n) @ m^T    -- two scalars per batch per step.
// One wave per batch: h[512] lives in registers (16 floats/lane), m/n columns
// register-resident. Only cross-lane op per step: 5-level shfl_xor reduction.
// No LDS, no barriers -> minimal per-step latency on the serial chain.
// ---------------------------------------------------------------------------
__global__ __launch_bounds__(32)
void lrnn_scan(const float* __restrict__ XP,   // [NB, NT, NH]
               const float* __restrict__ M,    // [NH, 2]
               const float* __restrict__ N,    // [NH, 2]
               float* __restrict__ OUT) {      // [NB, NT, NH]
  const int b    = blockIdx.x;
  const int lane = threadIdx.x;
  const int j0   = lane * 16;                  // this lane owns h[j0 .. j0+15]

  float h[16], m0v[16], m1v[16], n0v[16], n1v[16];
#pragma unroll
  for (int i = 0; i < 16; ++i) {
    v2f mv = *(const v2f*)(M + (size_t)(j0 + i) * 2);
    v2f nv = *(const v2f*)(N + (size_t)(j0 + i) * 2);
    m0v[i] = mv.x; m1v[i] = mv.y;
    n0v[i] = nv.x; n1v[i] = nv.y;
    h[i] = 0.0f;
  }

  const size_t base = (size_t)b * NT * NH + j0;
  for (int t = 0; t < NT; ++t) {
    const float* xp  = XP  + base + (size_t)t * NH;
    float*       out = OUT + base + (size_t)t * NH;

    // stream this step's input projection (4 x b128)
    v4f xpv[4];
#pragma unroll
    for (int q = 0; q < 4; ++q) xpv[q] = *(const v4f*)(xp + q * 4);

    // rank-2 contraction partials
    float s0 = 0.0f, s1 = 0.0f;
    float th[16];
#pragma unroll
    for (int i = 0; i < 16; ++i) {
      th[i] = tanhf(h[i]);
      s0 = fmaf(th[i], n0v[i], s0);
      s1 = fmaf(th[i], n1v[i], s1);
    }
    // wave32 butterfly reduction -> every lane holds full s0, s1
#pragma unroll
    for (int off = 16; off > 0; off >>= 1) {
      s0 += __shfl_xor(s0, off, 32);
      s1 += __shfl_xor(s1, off, 32);
    }

    // h += alpha * (-h + s0*m0 + s1*m1 + xp); emit hidden state
#pragma unroll
    for (int q = 0; q < 4; ++q) {
      v4f o;
#pragma unroll
      for (int e = 0; e < 4; ++e) {
        const int i = q * 4 + e;
        float r = fmaf(s0, m0v[i], fmaf(s1, m1v[i], xpv[q][e])) - h[i];
        h[i] = fmaf(ALPHA, r, h[i]);
        o[e] = h[i];
      }
      *(v4f*)(out + q * 4) = o;
    }
  }
}

// ---------------------------------------------------------------------------
extern "C" void kernel_launch(void* const* d_in, const int* in_sizes, int n_in,
                              void* d_out, int out_size, void* d_ws, size_t ws_size,
                              hipStream_t stream) {
  (void)in_sizes; (void)n_in; (void)out_size; (void)ws_size;
  const float* x = (const float*)d_in[0];   // [32, 2048, 128]
  const float* m = (const float*)d_in[1];   // [512, 2]
  const float* n = (const float*)d_in[2];   // [512, 2]
  const float* I = (const float*)d_in[3];   // [512, 128]
  float* out = (float*)d_out;               // [32, 2048, 512]
  float* xp  = (float*)d_ws;                // scratch: 32*2048*512*4 = 128 MB

  dim3 g1((NB * NT) / 16, NH / 16);         // 4096 x 32 tiles
  lrnn_xproj_wmma<<<g1, dim3(32), 0, stream>>>(x, I, xp);
  lrnn_scan<<<dim3(NB), dim3(32), 0, stream>>>(xp, m, n, out);
}